// GraphAttentionLayer_7026566496627
// MI455X (gfx1250) — compile-verified
//
#include <hip/hip_runtime.h>
#include <stdint.h>

typedef __attribute__((ext_vector_type(2))) float v2f;
typedef __attribute__((ext_vector_type(8))) float v8f;

#define GAT_ALPHA 0.2f

__device__ __forceinline__ float lrelu(float x) { return x > 0.f ? x : GAT_ALPHA * x; }

// Issue one 16KB chunk copy (global -> LDS) via async-to-LDS path.
// 256 threads x 4 x b128(16B) = 16384B. Each wave issues 4 wave-wide
// instructions (512B each), so per-wave ASYNCcnt grows by exactly 4.
__device__ __forceinline__ void async_copy_chunk16k(const float* gsrc,
                                                    uint32_t lds_u32, int tid) {
  uint64_t g = (uint64_t)(uintptr_t)gsrc + (uint64_t)(tid * 16);
  uint32_t l = lds_u32 + (uint32_t)(tid * 16);
#pragma unroll
  for (int i = 0; i < 4; ++i) {
    asm volatile("global_load_async_to_lds_b128 %0, %1, off"
                 :: "v"(l + (uint32_t)(i * 4096)),
                    "v"(g + (uint64_t)(i * 4096))
                 : "memory");
  }
}

// ---------------------------------------------------------------------------
// Kernel 0: Wt[k][n] = W[n][k]  (128x128, one-shot so Wh GEMM B-loads coalesce)
// ---------------------------------------------------------------------------
__global__ void __launch_bounds__(256) k_transpose(const float* __restrict__ W,
                                                   float* __restrict__ Wt) {
  int idx = blockIdx.x * 256 + threadIdx.x;   // 0..16383
  int n = idx >> 7, k = idx & 127;
  Wt[k * 128 + n] = W[idx];
}

// ---------------------------------------------------------------------------
// Kernel 1: Wh(16384x128) = h(16384x128) @ Wt(128x128) via V_WMMA_F32_16X16X4
// ---------------------------------------------------------------------------
__global__ void __launch_bounds__(256) k_wh(const float* __restrict__ h,
                                            const float* __restrict__ Wt,
                                            float* __restrict__ Wh) {
  int wave = threadIdx.x >> 5;
  int lane = threadIdx.x & 31;
  int rl   = lane & 15;       // row-in-tile (A) / col-in-tile (B,C)
  int half = lane >> 4;       // K-pair selector
  int m0   = (blockIdx.x * 8 + wave) * 16;

  v8f acc[8] = {};
  const float* hrow = h + (size_t)(m0 + rl) * 128 + half * 2;

  for (int k0 = 0; k0 < 128; k0 += 4) {
    v2f a = *(const v2f*)(hrow + k0);                       // A: K even/odd pair
    const float* B0 = Wt + (size_t)(k0 + half * 2) * 128 + rl;
    const float* B1 = B0 + 128;
#pragma unroll
    for (int t = 0; t < 8; ++t) {
      v2f bv; bv.x = B0[t * 16]; bv.y = B1[t * 16];
      acc[t] = __builtin_amdgcn_wmma_f32_16x16x4_f32(
          false, a, false, bv, (short)0, acc[t], false, false);
    }
  }
#pragma unroll
  for (int t = 0; t < 8; ++t)
#pragma unroll
    for (int v = 0; v < 8; ++v)
      Wh[(size_t)(m0 + v + half * 8) * 128 + t * 16 + rl] = acc[t][v];
}

// ---------------------------------------------------------------------------
// Kernel 2: s1[m] = Wh[m]·a1 + a_b ; s2[m] = Wh[m]·a2   (one wave per row)
// ---------------------------------------------------------------------------
__global__ void __launch_bounds__(256) k_s(const float* __restrict__ Wh,
                                           const float* __restrict__ a_w,
                                           const float* __restrict__ a_b,
                                           float* __restrict__ s1,
                                           float* __restrict__ s2) {
  int wave = threadIdx.x >> 5, lane = threadIdx.x & 31;
  int m = blockIdx.x * 8 + wave;                 // 0..16383
  float4 w  = ((const float4*)(Wh + (size_t)m * 128))[lane];
  float4 a1 = ((const float4*)a_w)[lane];
  float4 a2 = ((const float4*)(a_w + 128))[lane];
  float d1 = w.x * a1.x + w.y * a1.y + w.z * a1.z + w.w * a1.w;
  float d2 = w.x * a2.x + w.y * a2.y + w.z * a2.z + w.w * a2.w;
  for (int off = 16; off >= 1; off >>= 1) {
    d1 += __shfl_xor(d1, off);
    d2 += __shfl_xor(d2, off);
  }
  if (lane == 0) { s1[m] = d1 + a_b[0]; s2[m] = d2; }
}

// ---------------------------------------------------------------------------
// Kernel 3: fused masked-softmax(rank-1 scores) @ Wh, then elu.
// Wh is streamed through LDS with double-buffered async global->LDS copies
// (ASYNCcnt); 8 waves/block share each 16KB chunk; softmax weights are
// generated on the fly as the WMMA A-operand; normalize by 1/sum at the end.
// ---------------------------------------------------------------------------
__global__ void __launch_bounds__(256) k_attn(const float* __restrict__ Wh,
                                              const float* __restrict__ s1,
                                              const float* __restrict__ s2,
                                              float* __restrict__ out) {
  __shared__ float whbuf[2][32 * 128];   // 2 x 16KB double buffer (32 j-rows)
  __shared__ float s2s[1024];
  int tid  = threadIdx.x;
  int wave = tid >> 5, lane = tid & 31;
  int b    = blockIdx.x >> 3;
  int tile = (blockIdx.x & 7) * 8 + wave;        // 0..63 row-tiles in batch

  const float* WhB = Wh + (size_t)b * 1024 * 128;
  // LDS byte addresses: low 32 bits of the generic (shared) pointer are the
  // LDS offset (flat-aperture rule), which is what the async VDST wants.
  uint32_t buf_l0 = (uint32_t)(uintptr_t)&whbuf[0][0];
  uint32_t buf_l1 = (uint32_t)(uintptr_t)&whbuf[1][0];

  // stage s2 and kick off chunk 0 while we do the softmax-max prep
  ((float4*)s2s)[tid] = ((const float4*)(s2 + b * 1024))[tid];
  async_copy_chunk16k(WhB, buf_l0, tid);
  __syncthreads();

  // top-2 (max, argmax, 2nd max) of s2 for this batch (wave-redundant, cheap)
  float m1 = -3.0e38f, m2 = -3.0e38f; int i1 = -1;
  for (int j = lane; j < 1024; j += 32) {
    float v = s2s[j];
    if (v > m1) { m2 = m1; m1 = v; i1 = j; }
    else if (v > m2) { m2 = v; }
  }
  for (int off = 16; off >= 1; off >>= 1) {
    float om1 = __shfl_xor(m1, off);
    float om2 = __shfl_xor(m2, off);
    int   oi1 = __shfl_xor(i1, off);
    if (om1 > m1) { m2 = fmaxf(m1, om2); m1 = om1; i1 = oi1; }
    else          { m2 = fmaxf(m2, om1); }
  }

  int rl = lane & 15, half = lane >> 4;
  int r  = tile * 16 + rl;                       // row within batch
  float s1r  = s1[b * 1024 + r];                 // bias already folded in
  float Mrow = lrelu(s1r + ((r == i1) ? m2 : m1));   // true row max (diag masked)

  v8f acc[8] = {};
  float Ssum = 0.f;

  for (int c = 0; c < 32; ++c) {                 // 32 chunks x 32 j-rows
    if (c + 1 < 32) {
      async_copy_chunk16k(WhB + (size_t)(c + 1) * 32 * 128,
                          ((c + 1) & 1) ? buf_l1 : buf_l0, tid);
      asm volatile("s_wait_asynccnt 0x4" ::: "memory");   // chunk c landed
    } else {
      asm volatile("s_wait_asynccnt 0x0" ::: "memory");
    }
    __syncthreads();                             // chunk c visible to all waves

    const float* buf = (const float*)whbuf[c & 1];
#pragma unroll
    for (int kk = 0; kk < 32; kk += 4) {
      int jl = kk + half * 2;                    // local j-row in LDS chunk
      int jg = c * 32 + jl;                      // global j
      float e0 = lrelu(s1r + s2s[jg]);
      float e1 = lrelu(s1r + s2s[jg + 1]);
      float p0 = (jg     == r) ? 0.f : __expf(e0 - Mrow);  // diag -> weight 0
      float p1 = (jg + 1 == r) ? 0.f : __expf(e1 - Mrow);
      Ssum += p0 + p1;
      v2f a; a.x = p0; a.y = p1;
      const float* B0 = buf + jl * 128 + rl;
      const float* B1 = B0 + 128;
#pragma unroll
      for (int t = 0; t < 8; ++t) {
        v2f bv; bv.x = B0[t * 16]; bv.y = B1[t * 16];
        acc[t] = __builtin_amdgcn_wmma_f32_16x16x4_f32(
            false, a, false, bv, (short)0, acc[t], false, false);
      }
    }
    __syncthreads();                 // all waves done reading buf[c&1] before
  }                                  // it gets overwritten by chunk c+2

  Ssum += __shfl_xor(Ssum, 16);                  // halves of j-domain per row
  float inv = 1.f / Ssum;                        // keyed by lane%16 == row

  float* outB = out + (size_t)b * 1024 * 128;
#pragma unroll
  for (int v = 0; v < 8; ++v) {
    float invv = __shfl(inv, half ? (8 + v) : v);        // inv for C-row of VGPR v
    int row = tile * 16 + v + half * 8;
    float* orow = outB + (size_t)row * 128 + rl;
#pragma unroll
    for (int t = 0; t < 8; ++t) {
      float x = acc[t][v] * invv;
      orow[t * 16] = x > 0.f ? x : (__expf(x) - 1.f);    // elu
    }
  }
}

// ---------------------------------------------------------------------------
extern "C" void kernel_launch(void* const* d_in, const int* in_sizes, int n_in,
                              void* d_out, int out_size, void* d_ws, size_t ws_size,
                              hipStream_t stream) {
  const float* h   = (const float*)d_in[0];   // (16,1024,128)
  const float* W   = (const float*)d_in[1];   // (128,128)
  const float* a_w = (const float*)d_in[2];   // (256,)
  const float* a_b = (const float*)d_in[3];   // (1,)
  float* out = (float*)d_out;                 // (16,1024,128)

  float* ws = (float*)d_ws;
  float* Wt = ws;                 // 16384 floats
  float* Wh = Wt + 16384;         // 2097152 floats
  float* s1 = Wh + 2097152;       // 16384 floats
  float* s2 = s1 + 16384;         // 16384 floats  (~8.6 MB total)

  k_transpose<<<64,   256, 0, stream>>>(W, Wt);
  k_wh       <<<128,  256, 0, stream>>>(h, Wt, Wh);
  k_s        <<<2048, 256, 0, stream>>>(Wh, a_w, a_b, s1, s2);
  k_attn     <<<128,  256, 0, stream>>>(Wh, s1, s2, out);
}